// ContrastiveLabeledLoss_9328668967562
// MI455X (gfx1250) — compile-verified
//
#include <hip/hip_runtime.h>
#include <hip/hip_bf16.h>

// ---------------------------------------------------------------------------
// ContrastiveLabeledLoss for MI455X (gfx1250, wave32)
//   loss = sum_c  [count_c >= 2] * (1/(count_c-1)) * sum_{i: lab_i=c} ||x_i - x_anchor(c)||^2
// Memory-bound: one 256MB stream of `outputs`; anchors (<=1MB) live in L2.
// Squared row norms computed as the diagonal of Diff*Diff^T via
// V_WMMA_F32_16X16X4_F32 (A fragment == B fragment by layout symmetry).
// ---------------------------------------------------------------------------

#define DCOLS 256
#define CMAX  1024

typedef __attribute__((ext_vector_type(2))) float v2f;
typedef __attribute__((ext_vector_type(8))) float v8f;

__global__ void cll_init(int* __restrict__ counts, int* __restrict__ firstIdx,
                         float* __restrict__ loss, const int* __restrict__ nclsPtr) {
    int C = *nclsPtr; if (C > CMAX) C = CMAX;
    for (int i = blockIdx.x * blockDim.x + threadIdx.x; i < C;
         i += gridDim.x * blockDim.x) {
        counts[i]   = 0;
        firstIdx[i] = 0x7fffffff;
    }
    if (blockIdx.x == 0 && threadIdx.x == 0) *loss = 0.0f;
}

__global__ void cll_stats(const int* __restrict__ labels, int* __restrict__ counts,
                          int* __restrict__ firstIdx, int N,
                          const int* __restrict__ nclsPtr) {
    int C = *nclsPtr; if (C > CMAX) C = CMAX;
    for (int i = blockIdx.x * blockDim.x + threadIdx.x; i < N;
         i += gridDim.x * blockDim.x) {
        int lab = labels[i];
        if ((unsigned)lab < (unsigned)C) {
            atomicAdd(&counts[lab], 1);
            atomicMin(&firstIdx[lab], i);
        }
    }
}

__global__ void cll_weights(const int* __restrict__ counts,
                            float* __restrict__ invden,
                            const int* __restrict__ nclsPtr) {
    int C = *nclsPtr; if (C > CMAX) C = CMAX;
    for (int i = blockIdx.x * blockDim.x + threadIdx.x; i < C;
         i += gridDim.x * blockDim.x) {
        int c = counts[i];
        invden[i] = (c >= 2) ? (1.0f / (float)(c - 1)) : 0.0f;
    }
}

// One wave handles one 16-row tile. 8 waves (256 threads) per block.
__global__ __launch_bounds__(256) void cll_main(
    const float* __restrict__ outputs, const int* __restrict__ labels,
    const int* __restrict__ firstIdx, const float* __restrict__ invden,
    float* __restrict__ loss, int numTiles) {

    __shared__ float blockSum;
    const int tid = threadIdx.x;
    if (tid == 0) blockSum = 0.0f;
    __syncthreads();

    const int wave = tid >> 5;   // 0..7 (wave32)
    const int lane = tid & 31;
    const int tile = blockIdx.x * 8 + wave;   // wave-uniform

    float contrib = 0.0f;
    if (tile < numTiles) {                     // uniform per wave: EXEC stays all-1s
        const int row = tile * 16 + (lane & 15);
        int lab = labels[row];
        // clamp instead of branching so EXEC is never partial around WMMA
        lab = lab < 0 ? 0 : (lab >= CMAX ? CMAX - 1 : lab);
        const int   aidx = firstIdx[lab];
        const float w    = invden[lab];

        const float* __restrict__ rp = outputs + (size_t)row  * DCOLS;
        const float* __restrict__ ap = outputs + (size_t)aidx * DCOLS;
        // A 16x4 f32 layout: lanes 0-15 hold K = k..k+1, lanes 16-31 hold K = k+2..k+3
        const int coff = (lane >> 4) << 1;     // 0 or 2

        v8f acc = {};
#pragma unroll 8
        for (int k = 0; k < DCOLS; k += 4) {
            const int c0 = k + coff;                 // 8-byte aligned
            v2f x = *(const v2f*)(rp + c0);
            v2f a = *(const v2f*)(ap + c0);
            v2f d;
            d.x = x.x - a.x;
            d.y = x.y - a.y;
            // B (4x16) fragment layout mirrors A's, so the same registers feed
            // both operands: C += Diff * Diff^T. Diagonal = per-row ||diff||^2.
            acc = __builtin_amdgcn_wmma_f32_16x16x4_f32(
                false, d, false, d, (short)0, acc, false, false);
        }

        // Diagonal ownership (32-bit 16x16 C/D layout):
        //   lane r      (r=0..7)  holds (M=r,    N=r)    in acc[r]
        //   lane r+24   (r=0..7)  holds (M=r+8,  N=r+8)  in acc[r]
        // Both correspond to exactly the row this lane loaded.
        const bool isLow  = (lane < 8);
        const bool isHigh = (lane >= 24);
        const int  sel    = isLow ? lane : (lane - 24);
        float v = acc[0];
        v = (sel == 1) ? acc[1] : v;
        v = (sel == 2) ? acc[2] : v;
        v = (sel == 3) ? acc[3] : v;
        v = (sel == 4) ? acc[4] : v;
        v = (sel == 5) ? acc[5] : v;
        v = (sel == 6) ? acc[6] : v;
        v = (sel == 7) ? acc[7] : v;
        contrib = (isLow || isHigh) ? v * w : 0.0f;
    }

    atomicAdd(&blockSum, contrib);   // ds_add_f32
    __syncthreads();
    if (tid == 0) atomicAdd(loss, blockSum);
}

extern "C" void kernel_launch(void* const* d_in, const int* in_sizes, int n_in,
                              void* d_out, int out_size, void* d_ws, size_t ws_size,
                              hipStream_t stream) {
    const float* outputs = (const float*)d_in[0];
    const int*   labels  = (const int*)d_in[1];   // JAX default x64-off -> int32
    const int*   ncls    = (const int*)d_in[2];   // scalar (==1024)
    float*       loss    = (float*)d_out;

    const int N = in_sizes[1];                    // 262144 rows
    const int numTiles = N / 16;                  // rows are a multiple of 16

    int*   counts   = (int*)d_ws;                 // [CMAX]
    int*   firstIdx = counts + CMAX;              // [CMAX]
    float* invden   = (float*)(firstIdx + CMAX);  // [CMAX]
    (void)ws_size; (void)n_in; (void)out_size;

    cll_init<<<8, 256, 0, stream>>>(counts, firstIdx, loss, ncls);

    int statBlocks = (N + 255) / 256;
    if (statBlocks > 1024) statBlocks = 1024;
    cll_stats<<<statBlocks, 256, 0, stream>>>(labels, counts, firstIdx, N, ncls);

    cll_weights<<<8, 256, 0, stream>>>(counts, invden, ncls);

    const int mainBlocks = (numTiles + 7) / 8;    // 8 waves/block, 1 tile/wave
    cll_main<<<mainBlocks, 256, 0, stream>>>(outputs, labels, firstIdx, invden,
                                             loss, numTiles);
}